// LindBladEvolve_88734024336031
// MI455X (gfx1250) — compile-verified
//
#include <hip/hip_runtime.h>
#include <hip/hip_bf16.h>

// ---------------------------------------------------------------------------
// Lindblad control evolution for MI455X (gfx1250, wave32, WMMA).
//   D = 256 (complex), N_TS = 128, expm via scaling(2^6)+Taylor(12, Horner).
//   All heavy math = batched complex f32 GEMM on V_WMMA_F32_16X16X4_F32.
//   A-tiles staged global->LDS via async copy (ASYNCcnt) when available.
// ---------------------------------------------------------------------------

#define DD      256
#define NTS     128
#define NCTRL   2
#define TILE_M  64
#define TILE_N  64
#define TILE_K  32
#define BK_PAD  34   // padded row length (floats) for transposed B tiles

typedef __attribute__((ext_vector_type(2))) float v2f;
typedef __attribute__((ext_vector_type(8))) float v8f;
typedef __attribute__((ext_vector_type(4))) int   v4i;

#if defined(__has_builtin)
#if __has_builtin(__builtin_amdgcn_global_load_async_to_lds_b128) && \
    __has_builtin(__builtin_amdgcn_s_wait_asynccnt)
#define USE_ASYNC_COPY 1
#endif
#endif

#if defined(USE_ASYNC_COPY)
typedef __attribute__((address_space(1))) v4i* gas_ptr;   // global int4*
typedef __attribute__((address_space(3))) v4i* las_ptr;   // LDS int4*
#endif

static __device__ __forceinline__ v8f wmma_f32(v2f a, v2f b, v8f c) {
    // (neg_a, A, neg_b, B, c_mod, C, reuse_a, reuse_b); f32 WMMA has no A/B neg
    return __builtin_amdgcn_wmma_f32_16x16x4_f32(false, a, false, b,
                                                 (short)0, c, false, false);
}

// ---------------------------------------------------------------------------
// A_k = (dt/2^6) * [dyn + sum_j cr[k,j]*(G0+G1)_j + i*ci[k,j]*(G0-G1)_j]
// layout: A[ts][plane(re=0,im=1)][row][col]
// ---------------------------------------------------------------------------
__global__ __launch_bounds__(DD)
void build_A(const float* __restrict__ cr, const float* __restrict__ ci,
             const float* __restrict__ dyn_re, const float* __restrict__ dyn_im,
             const float* __restrict__ Gr, const float* __restrict__ Gi,
             float* __restrict__ Aout, float scale)
{
    const int ts  = blockIdx.y;
    const int row = blockIdx.x;
    const int col = threadIdx.x;
    const int e   = row * DD + col;

    float are = dyn_re[e];
    float aim = dyn_im[e];
#pragma unroll
    for (int j = 0; j < NCTRL; ++j) {
        const float g0r = Gr[(size_t)(j * 2 + 0) * DD * DD + e];
        const float g1r = Gr[(size_t)(j * 2 + 1) * DD * DD + e];
        const float g0i = Gi[(size_t)(j * 2 + 0) * DD * DD + e];
        const float g1i = Gi[(size_t)(j * 2 + 1) * DD * DD + e];
        const float c_r = cr[ts * NCTRL + j];
        const float c_i = ci[ts * NCTRL + j];
        // cr*(Gp_re + i Gp_im) + i*ci*(Gm_re + i Gm_im)
        are += c_r * (g0r + g1r) - c_i * (g0i - g1i);
        aim += c_r * (g0i + g1i) + c_i * (g0r - g1r);
    }
    const size_t o = (size_t)ts * 2 * DD * DD;
    Aout[o + e]           = are * scale;
    Aout[o + DD * DD + e] = aim * scale;
}

// R0 = I + A * (1/12)   (first Horner step, elementwise)
__global__ __launch_bounds__(DD)
void init_R(const float* __restrict__ A, float* __restrict__ R, float inv_k)
{
    const int ts  = blockIdx.y;
    const int row = blockIdx.x;
    const int col = threadIdx.x;
    const int e   = row * DD + col;
    const size_t o = (size_t)ts * 2 * DD * DD;
    R[o + e]           = A[o + e] * inv_k + ((row == col) ? 1.0f : 0.0f);
    R[o + DD * DD + e] = A[o + DD * DD + e] * inv_k;
}

// ---------------------------------------------------------------------------
// Batched complex GEMM:  C = alpha * (A @ B) (+ I if add_identity)
// Per block: 64x64 C tile; 8 waves each own a 16x32 strip (2 n-tiles).
// Complex via 3 accumulators per tile: RR=Are*Bre, II=Aim*Bim, MIX=cross.
// A tiles: async global->LDS (K contiguous). B tiles: staged transposed so
// every WMMA fragment (A and B) is one ds_load_b64.
// ---------------------------------------------------------------------------
__global__ __launch_bounds__(256)
void zgemm_batched(const float* __restrict__ A, const float* __restrict__ B,
                   float* __restrict__ C, float alpha, int add_identity)
{
    const int batch = blockIdx.y;
    const int tile  = blockIdx.x;            // 0..15
    const int m0 = (tile >> 2) * TILE_M;
    const int n0 = (tile & 3)  * TILE_N;

    const int tid  = threadIdx.x;
    const int lane = tid & 31;
    const int wave = tid >> 5;               // 0..7
    const int wm   = wave & 3;               // row strip 0..3  (16 rows each)
    const int wn   = wave >> 2;              // col half 0..1   (32 cols each)

    const size_t bofs = (size_t)batch * 2 * DD * DD;
    const float* __restrict__ Are = A + bofs;
    const float* __restrict__ Aim = Are + DD * DD;
    const float* __restrict__ Bre = B + bofs;
    const float* __restrict__ Bim = Bre + DD * DD;

    __shared__ __align__(16) float sAre[TILE_M][TILE_K];
    __shared__ __align__(16) float sAim[TILE_M][TILE_K];
    __shared__ __align__(16) float sBreT[TILE_N][BK_PAD];  // [n][k], K contiguous
    __shared__ __align__(16) float sBimT[TILE_N][BK_PAD];

    v8f accRR[2] = {};
    v8f accII[2] = {};
    v8f accMX[2] = {};

    const int am    = lane & 15;             // M (or N) lane index
    const int khalf = (lane >> 4) << 1;      // 0 or 2: K sub-pair select

    for (int kb = 0; kb < DD; kb += TILE_K) {
        // ---- A tiles: 64x32 per plane, 512 x 16B; async copy if available --
#pragma unroll
        for (int it = 0; it < 2; ++it) {
            const int li = tid + it * 256;
            const int r  = li >> 3;
            const int c4 = (li & 7) * 4;
            const float* gre = Are + (size_t)(m0 + r) * DD + kb + c4;
            const float* gim = Aim + (size_t)(m0 + r) * DD + kb + c4;
#if defined(USE_ASYNC_COPY)
            __builtin_amdgcn_global_load_async_to_lds_b128(
                (gas_ptr)gre, (las_ptr)&sAre[r][c4], 0, 0);
            __builtin_amdgcn_global_load_async_to_lds_b128(
                (gas_ptr)gim, (las_ptr)&sAim[r][c4], 0, 0);
#else
            *reinterpret_cast<float4*>(&sAre[r][c4]) =
                *reinterpret_cast<const float4*>(gre);
            *reinterpret_cast<float4*>(&sAim[r][c4]) =
                *reinterpret_cast<const float4*>(gim);
#endif
        }
        // ---- B tiles: 32x64 per plane, stored transposed ([n][k]) ---------
#pragma unroll
        for (int it = 0; it < 2; ++it) {
            const int li  = tid + it * 256;
            const int rb  = li >> 4;           // k row 0..31
            const int cb4 = (li & 15) * 4;     // n base 0..60
            const float4 vb_re = *reinterpret_cast<const float4*>(
                Bre + (size_t)(kb + rb) * DD + n0 + cb4);
            const float4 vb_im = *reinterpret_cast<const float4*>(
                Bim + (size_t)(kb + rb) * DD + n0 + cb4);
            sBreT[cb4 + 0][rb] = vb_re.x;
            sBreT[cb4 + 1][rb] = vb_re.y;
            sBreT[cb4 + 2][rb] = vb_re.z;
            sBreT[cb4 + 3][rb] = vb_re.w;
            sBimT[cb4 + 0][rb] = vb_im.x;
            sBimT[cb4 + 1][rb] = vb_im.y;
            sBimT[cb4 + 2][rb] = vb_im.z;
            sBimT[cb4 + 3][rb] = vb_im.w;
        }
#if defined(USE_ASYNC_COPY)
        __builtin_amdgcn_s_wait_asynccnt(0);
#endif
        __syncthreads();

        // ---- 8 x (K=4) WMMA steps; every fragment is one ds_load_b64 ------
#pragma unroll
        for (int kk = 0; kk < TILE_K; kk += 4) {
            const int krow = kk + khalf;       // K = krow, krow+1 per ISA layout
            const v2f a_re = *reinterpret_cast<const v2f*>(&sAre[wm * 16 + am][krow]);
            const v2f a_im = *reinterpret_cast<const v2f*>(&sAim[wm * 16 + am][krow]);
#pragma unroll
            for (int nt = 0; nt < 2; ++nt) {
                const int bn = wn * 32 + nt * 16 + am;
                const v2f b_re = *reinterpret_cast<const v2f*>(&sBreT[bn][krow]);
                const v2f b_im = *reinterpret_cast<const v2f*>(&sBimT[bn][krow]);
                accRR[nt] = wmma_f32(a_re, b_re, accRR[nt]);
                accII[nt] = wmma_f32(a_im, b_im, accII[nt]);
                accMX[nt] = wmma_f32(a_re, b_im, accMX[nt]);
                accMX[nt] = wmma_f32(a_im, b_re, accMX[nt]);
            }
        }
        __syncthreads();
    }

    // ---- epilogue: complex combine + alpha + optional identity ----
    float* __restrict__ Cre = C + bofs;
    float* __restrict__ Cim = Cre + DD * DD;
    const int lhalf = lane >> 4;
    const int lcol  = lane & 15;
#pragma unroll
    for (int nt = 0; nt < 2; ++nt) {
#pragma unroll
        for (int v = 0; v < 8; ++v) {
            const int m = m0 + wm * 16 + v + 8 * lhalf;   // global row
            const int n = n0 + wn * 32 + nt * 16 + lcol;  // global col
            float cre = (accRR[nt][v] - accII[nt][v]) * alpha;
            float cim = accMX[nt][v] * alpha;
            if (add_identity && (m == n)) cre += 1.0f;
            Cre[(size_t)m * DD + n] = cre;
            Cim[(size_t)m * DD + n] = cim;
        }
    }
}

// ---------------------------------------------------------------------------
// Sequential scan: s <- (P_k @ s) / trace(vec^-1(s)),  k = 0..127
// Single block of 256 threads (one row each). Then fid_err = ||sT - tgt||_F.
// out: [0..511] = interleaved (re,im) of sT, [512] = fid_err.
// ---------------------------------------------------------------------------
__global__ __launch_bounds__(DD)
void propagate(const float* __restrict__ P,
               const float* __restrict__ s0_re, const float* __restrict__ s0_im,
               const float* __restrict__ tgt_re, const float* __restrict__ tgt_im,
               float* __restrict__ out)
{
    __shared__ float sre[DD], sim[DD], nre[DD], nim[DD];
    __shared__ float red[DD];
    __shared__ float tr_re_s, tr_im_s;

    const int t = threadIdx.x;
    sre[t] = s0_re[t];
    sim[t] = s0_im[t];
    __syncthreads();

    for (int k = 0; k < NTS; ++k) {
        const float* __restrict__ Pre = P + (size_t)k * 2 * DD * DD + (size_t)t * DD;
        const float* __restrict__ Pim = Pre + DD * DD;
        float ar = 0.0f, ai = 0.0f;
        for (int c = 0; c < DD; ++c) {
            const float pr = Pre[c], pi = Pim[c];
            const float xr = sre[c], xi = sim[c];
            ar += pr * xr - pi * xi;
            ai += pr * xi + pi * xr;
        }
        nre[t] = ar;
        nim[t] = ai;
        __syncthreads();
        if (t == 0) {
            float trr = 0.0f, tri = 0.0f;
#pragma unroll
            for (int i = 0; i < 16; ++i) {   // diag_idx = i*(DIM+1), DIM=16
                trr += nre[i * 17];
                tri += nim[i * 17];
            }
            const float inv = 1.0f / (trr * trr + tri * tri);
            tr_re_s =  trr * inv;            // conj(trace) / |trace|^2
            tr_im_s = -tri * inv;
        }
        __syncthreads();
        const float dr = tr_re_s, di = tr_im_s;
        sre[t] = nre[t] * dr - nim[t] * di;
        sim[t] = nre[t] * di + nim[t] * dr;
        __syncthreads();
    }

    // Frobenius norm of (sT - tgt) over the complex 256-vector
    const float er = sre[t] - tgt_re[t];
    const float ei = sim[t] - tgt_im[t];
    red[t] = er * er + ei * ei;
    __syncthreads();
    for (int s = 128; s > 0; s >>= 1) {
        if (t < s) red[t] += red[t + s];
        __syncthreads();
    }
    out[2 * t]     = sre[t];
    out[2 * t + 1] = sim[t];
    if (t == 0) out[512] = sqrtf(red[0]);
}

// ---------------------------------------------------------------------------
extern "C" void kernel_launch(void* const* d_in, const int* in_sizes, int n_in,
                              void* d_out, int out_size, void* d_ws, size_t ws_size,
                              hipStream_t stream)
{
    const float* ctrls_real = (const float*)d_in[0];   // [128,2]
    const float* ctrls_im   = (const float*)d_in[1];   // [128,2]
    const float* dyn_re     = (const float*)d_in[2];   // [256,256]
    const float* dyn_im     = (const float*)d_in[3];
    const float* Gr         = (const float*)d_in[4];   // [2,2,256,256]
    const float* Gi         = (const float*)d_in[5];
    const float* init_re    = (const float*)d_in[6];   // [256]
    const float* init_im    = (const float*)d_in[7];
    const float* tgt_re     = (const float*)d_in[8];
    const float* tgt_im     = (const float*)d_in[9];

    const size_t plane = (size_t)NTS * 2 * DD * DD;    // floats per buffer
    float* Abuf = (float*)d_ws;                        // 64 MB
    float* P0   = Abuf + plane;                        // 64 MB
    float* P1   = P0 + plane;                          // 64 MB

    const float scale = 0.02f / 64.0f;                 // DT / 2^SQUARINGS

    build_A<<<dim3(DD, NTS), DD, 0, stream>>>(ctrls_real, ctrls_im,
                                              dyn_re, dyn_im, Gr, Gi,
                                              Abuf, scale);

    // Horner: R = I + A/12 done elementwise, then 11 GEMM steps
    init_R<<<dim3(DD, NTS), DD, 0, stream>>>(Abuf, P0, 1.0f / 12.0f);

    float* cur = P0;
    float* nxt = P1;
    for (int k = 11; k >= 1; --k) {
        zgemm_batched<<<dim3(16, NTS), 256, 0, stream>>>(
            Abuf, cur, nxt, 1.0f / (float)k, /*add_identity=*/1);
        float* tmp = cur; cur = nxt; nxt = tmp;
    }
    // 6 squarings
    for (int i = 0; i < 6; ++i) {
        zgemm_batched<<<dim3(16, NTS), 256, 0, stream>>>(
            cur, cur, nxt, 1.0f, /*add_identity=*/0);
        float* tmp = cur; cur = nxt; nxt = tmp;
    }

    propagate<<<1, DD, 0, stream>>>(cur, init_re, init_im, tgt_re, tgt_im,
                                    (float*)d_out);
}